// GCN_28501402976665
// MI455X (gfx1250) — compile-verified
//
#include <hip/hip_runtime.h>

typedef __attribute__((ext_vector_type(2))) float v2f;
typedef __attribute__((ext_vector_type(8))) float v8f;

#define NNODES 50000

// ---------- hardware f32 atomic add (no return, L2 RMW) ----------
__device__ __forceinline__ void atom_add_f32(float* p, float v) {
    asm volatile("global_atomic_add_f32 %0, %1, off" :: "v"(p), "v"(v) : "memory");
}

// ---------- init: deg=1 (self loop), zero agg1, zero out ----------
__global__ __launch_bounds__(256) void init_kernel(float* __restrict__ deg,
                                                   float* __restrict__ agg1,
                                                   float* __restrict__ out) {
    long long idx = (long long)blockIdx.x * 256 + threadIdx.x;
    if (idx < (long long)NNODES * 128) agg1[idx] = 0.0f;
    if (idx < (long long)NNODES * 64)  out[idx]  = 0.0f;
    if (idx < NNODES)                  deg[idx]  = 1.0f;
}

// ---------- degree count over edges ----------
__global__ __launch_bounds__(256) void deg_kernel(const int* __restrict__ dst,
                                                  float* __restrict__ deg, int E) {
    int e = blockIdx.x * 256 + threadIdx.x;
    if (e < E) atom_add_f32(&deg[dst[e]], 1.0f);
}

// ---------- deg -> rsqrt(deg) ----------
__global__ __launch_bounds__(256) void dinv_kernel(float* __restrict__ deg) {
    int i = blockIdx.x * 256 + threadIdx.x;
    if (i < NNODES) deg[i] = rsqrtf(deg[i]);   // deg >= 1 always
}

// ---------- WMMA f32 GEMM: C[M,NF] = A[M,K] * B[K,NF], wave per 16-row stripe ----------
template <int K, int NF>
__global__ __launch_bounds__(256) void gemm_wmma(const float* __restrict__ A,
                                                 const float* __restrict__ B,
                                                 float* __restrict__ C, int M) {
    const int lane  = threadIdx.x & 31;
    const int wave  = blockIdx.x * 8 + (threadIdx.x >> 5);
    const int mBase = wave * 16;
    if (mBase >= M) return;                 // uniform per wave; EXEC stays all-1s
    const int half = lane >> 4;             // 0: K+0/K+1 rows, 1: K+2/K+3 rows
    const int l16  = lane & 15;
    constexpr int NT = NF / 16;

    v8f acc[NT] = {};

    const float* aRow = A + (size_t)(mBase + l16) * K + 2 * half;
    for (int k = 0; k < K; k += 4) {
        v2f a = *(const v2f*)(aRow + k);    // A[m][k+2h], A[m][k+2h+1]
        const float* bPtr = B + (size_t)(k + 2 * half) * NF + l16;
#pragma unroll
        for (int t = 0; t < NT; ++t) {
            v2f b;
            b.x = bPtr[t * 16];             // B[k+2h  ][n]
            b.y = bPtr[t * 16 + NF];        // B[k+2h+1][n]
            acc[t] = __builtin_amdgcn_wmma_f32_16x16x4_f32(
                false, a, false, b, (short)0, acc[t], false, false);
        }
    }
    // C/D layout: VGPR r, lanes 0-15 -> row mBase+r, lanes 16-31 -> row mBase+8+r
#pragma unroll
    for (int t = 0; t < NT; ++t) {
        float* cPtr = C + (size_t)(mBase + half * 8) * NF + t * 16 + l16;
#pragma unroll
        for (int r = 0; r < 8; ++r) cPtr[(size_t)r * NF] = acc[t][r];
    }
}

// ---------- edge scatter: agg[d] += h[s] * dinv[s]*dinv[d], 4 feats/thread ----------
template <int F>
__global__ __launch_bounds__(256) void edge_agg(const int* __restrict__ src,
                                                const int* __restrict__ dst,
                                                const float* __restrict__ dinv,
                                                const float* __restrict__ h,
                                                float* __restrict__ agg, int E) {
    constexpr int CH = F / 4;
    long long tid = (long long)blockIdx.x * 256 + threadIdx.x;
    if (tid >= (long long)E * CH) return;
    int e = (int)(tid / CH);
    int c = (int)(tid & (CH - 1));
    int s = src[e], d = dst[e];
    float w = dinv[s] * dinv[d];
    float4 hv = ((const float4*)(h + (size_t)s * F))[c];
    float* ap = agg + (size_t)d * F + c * 4;
    atom_add_f32(ap + 0, hv.x * w);
    atom_add_f32(ap + 1, hv.y * w);
    atom_add_f32(ap + 2, hv.z * w);
    atom_add_f32(ap + 3, hv.w * w);
}

// ---------- layer1 finalize: agg1 = relu(agg1 + h1*dinv^2 + b1) ----------
__global__ __launch_bounds__(256) void fin1_kernel(float* __restrict__ agg1,
                                                   const float* __restrict__ h1,
                                                   const float* __restrict__ dinv,
                                                   const float* __restrict__ b1) {
    long long idx = (long long)blockIdx.x * 256 + threadIdx.x;
    if (idx >= (long long)NNODES * 128) return;
    int f = (int)(idx & 127);
    int i = (int)(idx >> 7);
    float di = dinv[i];
    float v = agg1[idx] + h1[idx] * di * di + b1[f];
    agg1[idx] = fmaxf(v, 0.0f);
}

// ---------- layer2 finalize: out = out + h2*dinv^2 + b2 ----------
__global__ __launch_bounds__(256) void fin2_kernel(float* __restrict__ out,
                                                   const float* __restrict__ h2,
                                                   const float* __restrict__ dinv,
                                                   const float* __restrict__ b2) {
    long long idx = (long long)blockIdx.x * 256 + threadIdx.x;
    if (idx >= (long long)NNODES * 64) return;
    int f = (int)(idx & 63);
    int i = (int)(idx >> 6);
    float di = dinv[i];
    out[idx] = out[idx] + h2[idx] * di * di + b2[f];
}

extern "C" void kernel_launch(void* const* d_in, const int* in_sizes, int n_in,
                              void* d_out, int out_size, void* d_ws, size_t ws_size,
                              hipStream_t stream) {
    const float* x  = (const float*)d_in[0];
    const int*   ei = (const int*)d_in[1];     // [2, E]
    const float* W1 = (const float*)d_in[2];
    const float* b1 = (const float*)d_in[3];
    const float* W2 = (const float*)d_in[4];
    const float* b2 = (const float*)d_in[5];
    float* out = (float*)d_out;

    const int N = NNODES;
    const int E = in_sizes[1] / 2;
    const int* src = ei;
    const int* dst = ei + E;

    // workspace layout
    float* deg  = (float*)d_ws;                 // N floats (becomes dinv)
    float* h1   = deg + N;                      // N*128
    float* agg1 = h1 + (size_t)N * 128;         // N*128
    float* h2   = agg1 + (size_t)N * 128;       // N*64

    const int BT = 256;
    long long n128 = (long long)N * 128;
    long long n64  = (long long)N * 64;

    // 1) init deg / zero agg1 / zero out
    init_kernel<<<(int)((n128 + BT - 1) / BT), BT, 0, stream>>>(deg, agg1, out);
    // 2) degree count (incoming edges; self loop already in init)
    deg_kernel<<<(E + BT - 1) / BT, BT, 0, stream>>>(dst, deg, E);
    // 3) dinv = rsqrt(deg)
    dinv_kernel<<<(N + BT - 1) / BT, BT, 0, stream>>>(deg);
    // 4) h1 = x @ W1  (M=50000, K=256, NF=128) -> 3125 stripes, 8 waves/block
    gemm_wmma<256, 128><<<(N / 16 + 7) / 8, BT, 0, stream>>>(x, W1, h1, N);
    // 5) scatter layer 1
    {
        long long tot = (long long)E * 32;
        edge_agg<128><<<(int)((tot + BT - 1) / BT), BT, 0, stream>>>(src, dst, deg, h1, agg1, E);
    }
    // 6) self loop + bias + relu
    fin1_kernel<<<(int)((n128 + BT - 1) / BT), BT, 0, stream>>>(agg1, h1, deg, b1);
    // 7) h2 = relu_out @ W2  (K=128, NF=64)
    gemm_wmma<128, 64><<<(N / 16 + 7) / 8, BT, 0, stream>>>(agg1, W2, h2, N);
    // 8) scatter layer 2 directly into out
    {
        long long tot = (long long)E * 16;
        edge_agg<64><<<(int)((tot + BT - 1) / BT), BT, 0, stream>>>(src, dst, deg, h2, out, E);
    }
    // 9) self loop + bias
    fin2_kernel<<<(int)((n64 + BT - 1) / BT), BT, 0, stream>>>(out, h2, deg, b2);
}